// DataAugmentation_4827543240891
// MI455X (gfx1250) — compile-verified
//
#include <hip/hip_runtime.h>
#include <hip/hip_bf16.h>
#include <stdint.h>

// Problem constants (fixed by reference setup_inputs)
#define B_   64
#define C_   3
#define HM   512
#define WMM  513   // width incl. meta column
#define WM   512   // usable width
#define OSZ  224

// ---------------------------------------------------------------------------
// JAX threefry2x32 (matches jax._src.prng.threefry2x32 exactly)
// ---------------------------------------------------------------------------
__device__ __forceinline__ uint32_t rotl32(uint32_t v, int d) {
  return (v << d) | (v >> (32 - d));
}

__device__ __forceinline__ void threefry2x32(uint32_t k0, uint32_t k1,
                                             uint32_t x0, uint32_t x1,
                                             uint32_t& o0, uint32_t& o1) {
  uint32_t ks2 = k0 ^ k1 ^ 0x1BD11BDAu;
  x0 += k0; x1 += k1;
#define TF_R4(a,b,c,d)                                   \
  x0 += x1; x1 = rotl32(x1,(a)); x1 ^= x0;               \
  x0 += x1; x1 = rotl32(x1,(b)); x1 ^= x0;               \
  x0 += x1; x1 = rotl32(x1,(c)); x1 ^= x0;               \
  x0 += x1; x1 = rotl32(x1,(d)); x1 ^= x0;
  TF_R4(13,15,26,6);  x0 += k1;  x1 += ks2 + 1u;
  TF_R4(17,29,16,24); x0 += ks2; x1 += k0  + 2u;
  TF_R4(13,15,26,6);  x0 += k0;  x1 += k1  + 3u;
  TF_R4(17,29,16,24); x0 += k1;  x1 += ks2 + 4u;
  TF_R4(13,15,26,6);  x0 += ks2; x1 += k0  + 5u;
#undef TF_R4
  o0 = x0; o1 = x1;
}

// random_bits(key, 32, (64,)) element b  (size 64 -> halves [0..31],[32..63])
__device__ __forceinline__ uint32_t rbits64(uint32_t ka, uint32_t kb, int b) {
  uint32_t y0, y1;
  if (b < 32) { threefry2x32(ka, kb, (uint32_t)b, (uint32_t)(b + 32), y0, y1); return y0; }
  threefry2x32(ka, kb, (uint32_t)(b - 32), (uint32_t)b, y0, y1);
  return y1;
}

// bits -> uniform [0,1) exactly as jax.random.uniform
__device__ __forceinline__ float u01f(uint32_t bits) {
  uint32_t f = (bits >> 9) | 0x3f800000u;
  return __uint_as_float(f) - 1.0f;
}

// ---------------------------------------------------------------------------
// Kernel 1: per-image crop params + flip decision (JAX-PRNG-exact)
// ws layout per image: [i, j, h, w, flip, pad, pad, pad]
// ---------------------------------------------------------------------------
__global__ __launch_bounds__(64)
void DataAug_params_kernel(const float* __restrict__ x, int* __restrict__ ws) {
  int b = threadIdx.x;
  if (b >= B_) return;

  // key(42) = (0,42); split -> kflip=(A0,B0), kparams=(A1,B1)
  uint32_t A0, A1, B0, B1;
  threefry2x32(0u, 42u, 0u, 2u, A0, A1);
  threefry2x32(0u, 42u, 1u, 3u, B0, B1);

  float Hf = x[((size_t)(b * C_ + 0) * HM + 0) * WMM + (WMM - 1)];
  float Wf = x[((size_t)(b * C_ + 1) * HM + 0) * WMM + (WMM - 1)];

  int flip = (u01f(rbits64(A0, B0, b)) > 0.5f) ? 1 : 0;

  const float log_lo = (float)-0.22314355131420976;  // log(0.8) as f32
  const float log_hi = (float) 0.22314355131420976;  // log(1.25) as f32
  float area = Hf * Wf;

  float fi = -1.0f, fj = -1.0f, fh = -1.0f, fw = -1.0f;
  bool success = false;

  for (int t = 0; t < 10; ++t) {
    // kt = fold_in(kparams, t) = threefry(kparams, (0, t))
    uint32_t kt0, kt1;
    threefry2x32(A1, B1, 0u, (uint32_t)t, kt0, kt1);
    // split(kt,4): counts iota(8) -> halves [0..3],[4..7]
    uint32_t y0_[4], y1_[4];
    for (int jj = 0; jj < 4; ++jj)
      threefry2x32(kt0, kt1, (uint32_t)jj, (uint32_t)(jj + 4), y0_[jj], y1_[jj]);
    // k1=(y0_0,y0_1) k2=(y0_2,y0_3) k3=(y1_0,y1_1) k4=(y1_2,y1_3)
    float u1 = fmaxf(0.1f,   u01f(rbits64(y0_[0], y0_[1], b)) * 0.9f + 0.1f);
    float u2 = fmaxf(log_lo, u01f(rbits64(y0_[2], y0_[3], b)) * (log_hi - log_lo) + log_lo);
    float u3 = u01f(rbits64(y1_[0], y1_[1], b));
    float u4 = u01f(rbits64(y1_[2], y1_[3], b));

    float ta = area * u1;
    float aspect = expf(u2);
    float cw = rintf(sqrtf(ta * aspect));   // round-half-even like jnp.round
    float ch = rintf(sqrtf(ta / aspect));
    bool valid = (cw > 0.0f) && (cw <= Wf) && (ch > 0.0f) && (ch <= Hf) && !success;
    float max_i = fmaxf(Hf - ch + 1.0f, 1.0f);
    float max_j = fmaxf(Wf - cw + 1.0f, 1.0f);
    float ri = floorf(u3 * max_i);
    float rj = floorf(u4 * max_j);
    if (valid) { fh = ch; fw = cw; fi = ri; fj = rj; success = true; }
  }

  if (!success) {
    float in_ratio = Wf / Hf;
    float fbw = (in_ratio > 1.25f) ? rintf(Hf * 1.25f) : Wf;
    float fbh = (in_ratio < 0.8f)  ? rintf(Wf / 0.8f)  : Hf;
    fi = floorf((Hf - fbh) * 0.5f);
    fj = floorf((Wf - fbw) * 0.5f);
    fh = fbh; fw = fbw;
  }

  ws[b * 8 + 0] = (int)fi;
  ws[b * 8 + 1] = (int)fj;
  ws[b * 8 + 2] = (int)fh;
  ws[b * 8 + 3] = (int)fw;
  ws[b * 8 + 4] = flip;
}

// ---------------------------------------------------------------------------
// Kernel 2: flip + crop + bilinear resize.
// One block per (image, output row). Stage 2 source rows x 3 channels into
// LDS via gfx1250 async global->LDS loads (ASYNCcnt path), then interpolate
// from LDS. Each HBM byte of the crop window is fetched exactly once per row
// use; x0/x1 neighbor overlap is resolved in LDS.
// ---------------------------------------------------------------------------
__global__ __launch_bounds__(256)
void DataAug_resize_kernel(const float* __restrict__ x,
                           const int* __restrict__ ws,
                           float* __restrict__ out) {
  __shared__ float smem[6 * WM];   // (channel, row) x up-to-512 floats = 12 KB

  const int blk = blockIdx.x;
  const int b  = blk / OSZ;
  const int oy = blk - b * OSZ;
  const int tid = threadIdx.x;

  const int pi   = ws[b * 8 + 0];
  const int pj   = ws[b * 8 + 1];
  const int ph   = ws[b * 8 + 2];
  const int pw   = ws[b * 8 + 3];
  const int flip = ws[b * 8 + 4];

  const float hf = (float)ph;
  const float wf = (float)pw;

  // vertical coordinates (uniform across block)
  float ys  = fminf(fmaxf(((float)oy + 0.5f) * hf / 224.0f - 0.5f, 0.0f), hf - 1.0f);
  float y0f = floorf(ys);
  float wy  = ys - y0f;
  int ycap  = min(max(pi + ph - 1, 0), HM - 1);
  int y0    = min(max((int)y0f + pi, 0), HM - 1);
  int y1    = min(y0 + 1, ycap);
  const bool two_rows = (y1 != y0);

  // horizontal segment of (flipped-space) columns touched: [seg_lo, seg_hi]
  const int seg_lo = pj;                      // pj >= 0 always
  const int seg_hi = min(pj + pw, WM - 1);
  const int len    = seg_hi - seg_lo + 1;     // <= 512
  // original-array column where the staged segment starts
  // (flip: flipped col xc maps to original col (WMM-1) - xc; may touch the
  //  meta column 512, exactly as the reference's full-width flip does)
  const int g0 = flip ? ((WMM - 1) - seg_hi) : seg_lo;

  // LDS index = xbase + xsign * x  (folds flip out of the inner loop)
  const int xsign = flip ? -1 : 1;
  const int xbase = flip ? seg_hi : -seg_lo;

  // ---- async stage: (c in 0..2) x (y0[,y1]) segments into LDS ----
  const int nrows = two_rows ? 2 : 1;
  for (int c = 0; c < C_; ++c) {
    for (int r = 0; r < nrows; ++r) {
      const int yr = r ? y1 : y0;
      const float* src = x + (((size_t)(b * C_ + c) * HM + yr) * WMM + g0);
      uint32_t lds_base = (uint32_t)(uintptr_t)(&smem[(c * 2 + r) * WM]);
      for (int k = tid; k < len; k += 256) {
        uint32_t voff = (uint32_t)(k * 4);
        uint32_t ldsa = lds_base + (uint32_t)(k * 4);
        asm volatile("global_load_async_to_lds_b32 %0, %1, %2"
                     :
                     : "v"(ldsa), "v"(voff), "s"(src)
                     : "memory");
      }
    }
  }
  asm volatile("s_wait_asynccnt 0" ::: "memory");
  __syncthreads();

  // ---- bilinear from LDS: 3 channels x 224 pixels per block ----
  const int xcap = min(max(pj + pw - 1, 0), WM - 1);
  for (int p = tid; p < C_ * OSZ; p += 256) {
    const int c  = p / OSZ;
    const int ox = p - c * OSZ;
    float xs  = fminf(fmaxf(((float)ox + 0.5f) * wf / 224.0f - 0.5f, 0.0f), wf - 1.0f);
    float x0f = floorf(xs);
    float wx  = xs - x0f;
    int x0 = min(max((int)x0f + pj, 0), WM - 1);
    int x1 = min(x0 + 1, xcap);
    int i0 = xbase + xsign * x0;
    int i1 = xbase + xsign * x1;
    const float* r0 = &smem[(c * 2 + 0) * WM];
    const float* r1 = two_rows ? &smem[(c * 2 + 1) * WM] : r0;
    float s00 = r0[i0], s01 = r0[i1];
    float s10 = r1[i0], s11 = r1[i1];
    float top = (1.0f - wx) * s00 + wx * s01;
    float bot = (1.0f - wx) * s10 + wx * s11;
    float val = (1.0f - wy) * top + wy * bot;
    out[(((size_t)b * C_ + c) * OSZ + oy) * OSZ + ox] = val;
  }
}

// ---------------------------------------------------------------------------
extern "C" void kernel_launch(void* const* d_in, const int* in_sizes, int n_in,
                              void* d_out, int out_size, void* d_ws, size_t ws_size,
                              hipStream_t stream) {
  const float* x = (const float*)d_in[0];
  int* ws = (int*)d_ws;
  float* out = (float*)d_out;
  (void)in_sizes; (void)n_in; (void)out_size; (void)ws_size;

  DataAug_params_kernel<<<dim3(1), dim3(64), 0, stream>>>(x, ws);
  DataAug_resize_kernel<<<dim3(B_ * OSZ), dim3(256), 0, stream>>>(x, ws, out);
}